// Graph_Layer_44787918963018
// MI455X (gfx1250) — compile-verified
//
#include <hip/hip_runtime.h>

#define NROWS 8192
#define DDIM  256
#define EPSN  1e-12f
#define KSLABS 32   // slabs for the rank-8192 update (8192/32 = 256 rows each)

typedef __attribute__((ext_vector_type(2))) float v2f;
typedef __attribute__((ext_vector_type(8))) float v8f;

__device__ __forceinline__ float wave_sum(float v) {
#pragma unroll
    for (int o = 16; o > 0; o >>= 1) v += __shfl_xor(v, o, 32);
    return v;
}

__device__ __forceinline__ v8f wmma_f32(v2f a, v2f b, v8f c) {
    // V_WMMA_F32_16X16X4_F32 : D = A(16x4) * B(4x16) + C, full f32
    return __builtin_amdgcn_wmma_f32_16x16x4_f32(
        /*neg_a=*/false, a, /*neg_b=*/false, b,
        /*c_mod=*/(short)0, c, /*reuse_a=*/false, /*reuse_b=*/false);
}

// ---------------------------------------------------------------------------
// Kernel A: per-row inverse norm + deterministic partial column sums of xn.
// 128 blocks x 256 thr; each wave handles 8 rows (wave-per-row, float4 loads).
// ---------------------------------------------------------------------------
__global__ void __launch_bounds__(256)
k_norm_colsum(const float* __restrict__ x, float* __restrict__ inv,
              float* __restrict__ partial_s) {
    __shared__ __align__(16) float lds[8 * DDIM];
    const int lane  = threadIdx.x & 31;
    const int wave  = threadIdx.x >> 5;
    const int rbase = blockIdx.x * 64 + wave * 8;

    float c0x = 0.f, c0y = 0.f, c0z = 0.f, c0w = 0.f;
    float c1x = 0.f, c1y = 0.f, c1z = 0.f, c1w = 0.f;

#pragma unroll
    for (int r = 0; r < 8; ++r) {
        const int row = rbase + r;
        const float4 a = *(const float4*)(x + row * DDIM + lane * 4);
        const float4 b = *(const float4*)(x + row * DDIM + 128 + lane * 4);
        float ss = a.x*a.x + a.y*a.y + a.z*a.z + a.w*a.w
                 + b.x*b.x + b.y*b.y + b.z*b.z + b.w*b.w;
        ss = wave_sum(ss);
        const float iv = 1.0f / fmaxf(sqrtf(ss), EPSN);
        if (lane == 0) inv[row] = iv;
        c0x += iv * a.x; c0y += iv * a.y; c0z += iv * a.z; c0w += iv * a.w;
        c1x += iv * b.x; c1y += iv * b.y; c1z += iv * b.z; c1w += iv * b.w;
    }
    float4* l0 = (float4*)(lds + wave * DDIM + lane * 4);
    float4* l1 = (float4*)(lds + wave * DDIM + 128 + lane * 4);
    *l0 = make_float4(c0x, c0y, c0z, c0w);
    *l1 = make_float4(c1x, c1y, c1z, c1w);
    __syncthreads();

    float p = 0.f;
#pragma unroll
    for (int w = 0; w < 8; ++w) p += lds[w * DDIM + threadIdx.x];
    partial_s[blockIdx.x * DDIM + threadIdx.x] = p;   // each block writes its full row
}

// ---------------------------------------------------------------------------
// Kernel A2: s[t] = sum over 128 block partials (deterministic).
// ---------------------------------------------------------------------------
__global__ void __launch_bounds__(256)
k_colsum_reduce(const float* __restrict__ partial_s, float* __restrict__ s) {
    float acc = 0.f;
#pragma unroll 8
    for (int b = 0; b < 128; ++b) acc += partial_s[b * DDIM + threadIdx.x];
    s[threadIdx.x] = acc;
}

// ---------------------------------------------------------------------------
// Kernel B: coef_i = rsqrt(inv_i * (x_i . s)) * inv_i   (= d_i / ||x_i||)
// ---------------------------------------------------------------------------
__global__ void __launch_bounds__(256)
k_coef(const float* __restrict__ x, const float* __restrict__ inv,
       const float* __restrict__ s, float* __restrict__ coef) {
    __shared__ __align__(16) float ls[DDIM];
    ls[threadIdx.x] = s[threadIdx.x];
    __syncthreads();
    const int lane  = threadIdx.x & 31;
    const int wave  = threadIdx.x >> 5;
    const int rbase = blockIdx.x * 64 + wave * 8;
    const float4 sa = *(const float4*)(ls + lane * 4);
    const float4 sb = *(const float4*)(ls + 128 + lane * 4);
#pragma unroll
    for (int r = 0; r < 8; ++r) {
        const int row = rbase + r;
        const float4 a = *(const float4*)(x + row * DDIM + lane * 4);
        const float4 b = *(const float4*)(x + row * DDIM + 128 + lane * 4);
        float d = a.x*sa.x + a.y*sa.y + a.z*sa.z + a.w*sa.w
                + b.x*sb.x + b.y*sb.y + b.z*sb.z + b.w*sb.w;
        d = wave_sum(d);
        if (lane == 0) {
            const float iv = inv[row];
            coef[row] = rsqrtf(iv * d) * iv;
        }
    }
}

// ---------------------------------------------------------------------------
// Kernel C: Mpart[slab] = Z_slab^T @ X_slab  (Z = diag(coef) X), WMMA f32.
// grid (32 blocks x 32 slabs), 8 waves/block -> 256 waves cover all 16x16
// output tiles of the 256x256 result per slab. K per slab = 256.
// ---------------------------------------------------------------------------
__global__ void __launch_bounds__(256)
k_ztx(const float* __restrict__ x, const float* __restrict__ coef,
      float* __restrict__ Mpart) {
    const int lane = threadIdx.x & 31;
    const int wave = threadIdx.x >> 5;
    const int tile = blockIdx.x * 8 + wave;       // 0..255
    const int tm = (tile >> 4) * 16;              // output row block (Z cols)
    const int tn = (tile & 15) * 16;              // output col block (X cols)
    const int kstart = blockIdx.y * (NROWS / KSLABS);
    const int n  = lane & 15;                     // m for A, n for B
    const int kh = (lane >> 4) * 2;               // lane-half K offset (0 or 2)

    v8f acc = {};
#pragma unroll 4
    for (int kk = 0; kk < NROWS / KSLABS; kk += 4) {
        const int kg = kstart + kk + kh;
        v2f a, b;
        a.x = coef[kg]     * x[kg * DDIM + tm + n];        // A[m][kh+0] = Z^T
        a.y = coef[kg + 1] * x[(kg + 1) * DDIM + tm + n];  // A[m][kh+1]
        b.x = x[kg * DDIM + tn + n];                       // B[kh+0][n]
        b.y = x[(kg + 1) * DDIM + tn + n];                 // B[kh+1][n]
        acc = wmma_f32(a, b, acc);
    }
    float* out = Mpart + blockIdx.y * (DDIM * DDIM);
    const int row0 = tm + (lane >> 4) * 8;
#pragma unroll
    for (int r = 0; r < 8; ++r)
        out[(row0 + r) * DDIM + tn + n] = acc[r];
}

// ---------------------------------------------------------------------------
// Kernel D: M = sum over KSLABS partials (deterministic tree).
// ---------------------------------------------------------------------------
__global__ void __launch_bounds__(256)
k_reduce_m(const float* __restrict__ Mpart, float* __restrict__ M) {
    const int idx = blockIdx.x * 256 + threadIdx.x;
    float acc = 0.f;
#pragma unroll 8
    for (int p = 0; p < KSLABS; ++p) acc += Mpart[p * (DDIM * DDIM) + idx];
    M[idx] = acc;
}

// ---------------------------------------------------------------------------
// Kernel E: P = M @ W  (256x256x256), WMMA f32. 32 blocks x 8 waves = 256 tiles.
// ---------------------------------------------------------------------------
__global__ void __launch_bounds__(256)
k_mm256(const float* __restrict__ A, const float* __restrict__ B,
        float* __restrict__ C) {
    const int lane = threadIdx.x & 31;
    const int wave = threadIdx.x >> 5;
    const int tile = blockIdx.x * 8 + wave;
    const int tm = (tile >> 4) * 16;
    const int tn = (tile & 15) * 16;
    const int n  = lane & 15;
    const int kh = (lane >> 4) * 2;

    v8f acc = {};
#pragma unroll 4
    for (int kk = 0; kk < DDIM; kk += 4) {
        v2f a = *(const v2f*)(A + (tm + n) * DDIM + kk + kh);  // row-major A
        v2f b;
        b.x = B[(kk + kh) * DDIM + tn + n];
        b.y = B[(kk + kh + 1) * DDIM + tn + n];
        acc = wmma_f32(a, b, acc);
    }
    const int row0 = tm + (lane >> 4) * 8;
#pragma unroll
    for (int r = 0; r < 8; ++r)
        C[(row0 + r) * DDIM + tn + n] = acc[r];
}

// ---------------------------------------------------------------------------
// Kernel F: out = diag(coef) X @ P   [8192,256]. 1024 blocks x 8 waves =
// 8192 tiles (512 row-tiles x 16 col-tiles). coef folded into A on the fly.
// ---------------------------------------------------------------------------
__global__ void __launch_bounds__(256)
k_out(const float* __restrict__ x, const float* __restrict__ coef,
      const float* __restrict__ P, float* __restrict__ out) {
    const int lane = threadIdx.x & 31;
    const int wave = threadIdx.x >> 5;
    const int tile = blockIdx.x * 8 + wave;       // 0..8191
    const int tm = (tile >> 4) * 16;              // row tile base (0..8176)
    const int tn = (tile & 15) * 16;
    const int n  = lane & 15;
    const int kh = (lane >> 4) * 2;
    const int row = tm + n;                       // A-row handled by this lane
    const float cf = coef[row];

    v8f acc = {};
#pragma unroll 4
    for (int kk = 0; kk < DDIM; kk += 4) {
        v2f a = *(const v2f*)(x + row * DDIM + kk + kh);
        a.x *= cf; a.y *= cf;
        v2f b;
        b.x = P[(kk + kh) * DDIM + tn + n];
        b.y = P[(kk + kh + 1) * DDIM + tn + n];
        acc = wmma_f32(a, b, acc);
    }
    const int row0 = tm + (lane >> 4) * 8;
#pragma unroll
    for (int r = 0; r < 8; ++r)
        out[(row0 + r) * DDIM + tn + n] = acc[r];
}

// ---------------------------------------------------------------------------
extern "C" void kernel_launch(void* const* d_in, const int* in_sizes, int n_in,
                              void* d_out, int out_size, void* d_ws, size_t ws_size,
                              hipStream_t stream) {
    (void)in_sizes; (void)n_in; (void)out_size; (void)ws_size;
    const float* x = (const float*)d_in[0];   // [8192,256]
    const float* W = (const float*)d_in[1];   // [256,256]
    float* out = (float*)d_out;               // [8192,256]

    float* ws        = (float*)d_ws;
    float* partial_s = ws;                                  // 128*256
    float* s         = partial_s + 128 * DDIM;              // 256
    float* inv       = s + DDIM;                            // 8192
    float* coef      = inv + NROWS;                         // 8192
    float* Mpart     = coef + NROWS;                        // 32*256*256 (8 MB)
    float* M         = Mpart + KSLABS * DDIM * DDIM;        // 256*256
    float* P         = M + DDIM * DDIM;                     // 256*256

    k_norm_colsum <<<128, 256, 0, stream>>>(x, inv, partial_s);
    k_colsum_reduce<<<1,  256, 0, stream>>>(partial_s, s);
    k_coef        <<<128, 256, 0, stream>>>(x, inv, s, coef);
    k_ztx         <<<dim3(32, KSLABS), 256, 0, stream>>>(x, coef, Mpart);
    k_reduce_m    <<<256, 256, 0, stream>>>(Mpart, M);
    k_mm256       <<<32,  256, 0, stream>>>(M, W, P);
    k_out         <<<1024, 256, 0, stream>>>(x, coef, P, out);
}